// Extract_HyperSpherePrototypes_31542239822124
// MI455X (gfx1250) — compile-verified
//
#include <hip/hip_runtime.h>
#include <stdint.h>

typedef __attribute__((ext_vector_type(2))) float v2f;
typedef __attribute__((ext_vector_type(8))) float v8f;

#define BSZ   16
#define FCH   256
#define HW    16384          // 128*128
#define NPIX  (BSZ*HW)       // 262144 pixels
#define TILE  64             // pixels per LDS tile
#define NTILES (NPIX/TILE)   // 4096
#define LSTR  68             // padded LDS row stride (floats) -> conflict-free A reads
#define NCLS  19
#define CPAD  32             // padded class dim (2 WMMA N-tiles)
#define PROTO (FCH*CPAD)     // 8192 floats per partial
#define GRID  256
#define BLOCK 256

__global__ __launch_bounds__(BLOCK) void proto_zero_kernel(float* p, int n) {
    int i = blockIdx.x * BLOCK + threadIdx.x;
    if (i < n) p[i] = 0.0f;
}

// Fused: async-LDS double-buffered streaming + per-pixel L2-normalize +
// one-hot GEMM scatter-sum via fp32 WMMA.
__global__ __launch_bounds__(BLOCK) void proto_accum_kernel(
        const float* __restrict__ feat, const int* __restrict__ lab,
        float* __restrict__ parts, int nparts) {
    __shared__ float s_feat[2][FCH * LSTR];   // 2 x ~68 KB (double buffer)
    __shared__ float s_inv[TILE];
    __shared__ int   s_lab[TILE];

    const int tid   = threadIdx.x;
    const int lane  = tid & 31;
    const int wv    = tid >> 5;                 // wave id 0..7
    const int row   = lane & 15;                // M (channel) / N (class) row within tile
    const int khalf = (lane < 16) ? 0 : 2;      // ISA: lanes 0-15 hold K=0,1; 16-31 hold K=2,3

    const int lp = (tid & 15) << 2;             // pixel 0..60 step 4 (16B chunk)
    const int c0 = tid >> 4;                    // channel 0..15

    // Issue async global->LDS copies for one [256ch x 64px] tile (16 x b128 per lane).
    auto issue_tile = [&](int tile, int buf) {
        const int p0 = tile * TILE;
        const float* src = feat + (((size_t)(p0 >> 14) * FCH) << 14) + (p0 & (HW - 1)) + lp;
        #pragma unroll
        for (int cc = 0; cc < FCH; cc += 16) {
            const int c = cc + c0;
            const unsigned long long g =
                (unsigned long long)(uintptr_t)(src + ((size_t)c << 14));
            // low 32 bits of a flat shared pointer == LDS byte offset (aperture in high bits)
            const unsigned l = (unsigned)(uintptr_t)&s_feat[buf][c * LSTR + lp];
            asm volatile("global_load_async_to_lds_b128 %0, %1, off"
                         :: "v"(l), "v"(g) : "memory");
        }
    };

    v8f acc[2][2] = {};                         // [Mtile][Ntile] fp32 accumulators
    int cur = 0;

    issue_tile(blockIdx.x, 0);                  // prologue: tile 0 into buffer 0

    for (int tile = blockIdx.x; tile < NTILES; tile += gridDim.x) {
        // Wait for this wave's async loads into `cur`, then block-wide sync:
        // guarantees cur buffer fully populated AND everyone done reading cur^1.
        asm volatile("s_wait_asynccnt 0x0" ::: "memory");
        __syncthreads();

        const int nxt = tile + gridDim.x;       // uniform per block (4096 % 256 == 0)
        if (nxt < NTILES) issue_tile(nxt, cur ^ 1);   // overlap with compute below

        if (tid < TILE) s_lab[tid] = lab[tile * TILE + tid];

        const float* fb = &s_feat[cur][0];

        // ---- per-pixel inverse L2 norm (4 lanes per pixel + shfl_xor reduce) ----
        {
            const int p = tid >> 2;             // pixel 0..63
            const int q = tid & 3;              // channel quarter
            float s = 0.0f;
            #pragma unroll 8
            for (int c = q; c < FCH; c += 4) {
                float f = fb[c * LSTR + p];
                s += f * f;
            }
            s += __shfl_xor(s, 1, 32);
            s += __shfl_xor(s, 2, 32);
            if (q == 0) s_inv[p] = 1.0f / fmaxf(sqrtf(s), 1e-12f);
        }
        __syncthreads();

        // ---- WMMA: D[ch, cls] += (feat*inv)[ch, px] x one_hot[px, cls] ----
        #pragma unroll 4
        for (int k = 0; k < TILE; k += 4) {
            const int   kb = k + khalf;
            const float i0 = s_inv[kb];
            const float i1 = s_inv[kb + 1];
            const int   l0 = s_lab[kb];
            const int   l1 = s_lab[kb + 1];

            v2f A[2];
            #pragma unroll
            for (int mt = 0; mt < 2; ++mt) {
                const int c = (2 * wv + mt) * 16 + row;
                A[mt].x = fb[c * LSTR + kb]     * i0;
                A[mt].y = fb[c * LSTR + kb + 1] * i1;
            }
            #pragma unroll
            for (int nt = 0; nt < 2; ++nt) {
                const int cls = nt * 16 + row;  // B: N = lane%16, K mirrored like A
                v2f B;
                B.x = (l0 == cls) ? 1.0f : 0.0f;
                B.y = (l1 == cls) ? 1.0f : 0.0f;
                #pragma unroll
                for (int mt = 0; mt < 2; ++mt) {
                    acc[mt][nt] = __builtin_amdgcn_wmma_f32_16x16x4_f32(
                        false, A[mt], false, B, (short)0, acc[mt][nt], false, false);
                }
            }
        }
        cur ^= 1;
    }

    // ---- write per-block partial prototypes (uncontended when nparts==GRID) ----
    float* pb = parts + (size_t)(blockIdx.x % nparts) * PROTO;
    const int mofs = (lane < 16) ? 0 : 8;   // D layout: VGPR r -> M = r (+8 for hi lanes)
    const int ncol = lane & 15;
    #pragma unroll
    for (int mt = 0; mt < 2; ++mt)
        #pragma unroll
        for (int nt = 0; nt < 2; ++nt)
            #pragma unroll
            for (int r = 0; r < 8; ++r) {
                const int ch = (2 * wv + mt) * 16 + mofs + r;
                atomicAdd(&pb[ch * CPAD + nt * 16 + ncol], acc[mt][nt][r]);
            }
}

__global__ __launch_bounds__(BLOCK) void proto_reduce_kernel(float* parts, int nparts) {
    const int e = blockIdx.x * BLOCK + threadIdx.x;   // 32 blocks -> 8192 elements
    float s = parts[e];
    for (int k = 1; k < nparts; ++k) s += parts[(size_t)k * PROTO + e];
    parts[e] = s;
}

__global__ __launch_bounds__(BLOCK) void proto_final_kernel(
        const float* __restrict__ parts, float* __restrict__ out) {
    __shared__ float s_sq[8 * CPAD];
    __shared__ float s_inv[CPAD];
    const int tid = threadIdx.x;
    const int col = tid & 31;
    const int grp = tid >> 5;
    float s = 0.0f;
    for (int r = grp * 32; r < grp * 32 + 32; ++r) {
        float v = parts[r * CPAD + col];
        s += v * v;
    }
    s_sq[grp * CPAD + col] = s;
    __syncthreads();
    if (tid < CPAD) {
        float t = 0.0f;
        #pragma unroll
        for (int g = 0; g < 8; ++g) t += s_sq[g * CPAD + tid];
        s_inv[tid] = 1.0f / fmaxf(sqrtf(t), 1e-12f);
    }
    __syncthreads();
    for (int idx = tid; idx < FCH * NCLS; idx += BLOCK) {
        const int f = idx / NCLS;
        const int c = idx % NCLS;
        out[idx] = parts[f * CPAD + c] * s_inv[c];
    }
}

extern "C" void kernel_launch(void* const* d_in, const int* in_sizes, int n_in,
                              void* d_out, int out_size, void* d_ws, size_t ws_size,
                              hipStream_t stream) {
    (void)in_sizes; (void)n_in; (void)out_size;
    const float* feat   = (const float*)d_in[0];
    const int*   labels = (const int*)d_in[1];
    float*       out    = (float*)d_out;
    float*       parts  = (float*)d_ws;

    int maxparts = (int)(ws_size / (PROTO * sizeof(float)));
    int nparts = maxparts < 1 ? 1 : (maxparts > GRID ? GRID : maxparts);

    const int ztot = nparts * PROTO;
    proto_zero_kernel<<<(ztot + BLOCK - 1) / BLOCK, BLOCK, 0, stream>>>(parts, ztot);
    proto_accum_kernel<<<GRID, BLOCK, 0, stream>>>(feat, labels, parts, nparts);
    if (nparts > 1)
        proto_reduce_kernel<<<PROTO / BLOCK, BLOCK, 0, stream>>>(parts, nparts);
    proto_final_kernel<<<1, BLOCK, 0, stream>>>(parts, out);
}